// GATEdgeClassifier_40037685133537
// MI455X (gfx1250) — compile-verified
//
#include <hip/hip_runtime.h>
#include <hip/hip_bf16.h>
#include <math.h>

// ---------------------------------------------------------------------------
// GAT edge classifier for MI455X (gfx1250, wave32).
//   Layer1: GATConv(256 -> 4 heads x 64), relu
//   Layer2: GATConv(256 -> 1 head x 64)
//   Head:   concat(h[src], h[dst]) @ Wc + bc   -> [E, 4]
// Dense GEMMs use V_WMMA_F32_16X16X4_F32 (exact fp32 matrix path).
// Edge softmax/scatter is the memory-bound part: float4 gathers + f32 atomics.
// ---------------------------------------------------------------------------

#define NNODES 50000
#define NEDGES 800000
#define INCH   256
#define HID    64
#define HEADS  4
#define OUTCH  4
#define NEG_SLOPE 0.2f

typedef __attribute__((ext_vector_type(2))) float v2f;
typedef __attribute__((ext_vector_type(8))) float v8f;

// ---------------------------------------------------------------------------
// Generic fill
// ---------------------------------------------------------------------------
__global__ void fill_f32(float* __restrict__ p, float v, size_t n) {
  size_t i = (size_t)blockIdx.x * blockDim.x + threadIdx.x;
  size_t stride = (size_t)gridDim.x * blockDim.x;
  for (; i < n; i += stride) p[i] = v;
}

// ---------------------------------------------------------------------------
// WMMA fp32 GEMM: C[M,N] = A[M,K] @ B[K,N], row-major, M%16==N%16==K%4==0.
// One wave per 16x16 C tile; K-loop in steps of 4 via V_WMMA_F32_16X16X4_F32.
// A 16x4 layout: lanes 0-15 -> rows 0-15 holding K=k..k+1, lanes 16-31 hold
// K=k+2..k+3. B 4x16 is the mirrored layout (lane%16 = column).
// D layout: VGPR j -> row j (lanes 0-15) / row j+8 (lanes 16-31), col=lane%16.
// ---------------------------------------------------------------------------
__global__ void wmma_gemm_f32(const float* __restrict__ A,
                              const float* __restrict__ B,
                              float* __restrict__ C,
                              int M, int N, int K) {
  const int lane = threadIdx.x & 31;
  const int wave = (blockIdx.x * blockDim.x + threadIdx.x) >> 5;
  const int ntiles = N >> 4;
  const int mtiles = M >> 4;
  if (wave >= mtiles * ntiles) return;           // wave-uniform: EXEC stays full
  const int mt = wave / ntiles;
  const int nt = wave - mt * ntiles;

  const int half = lane >> 4;                    // 0: lanes 0-15, 1: lanes 16-31
  const int l16  = lane & 15;
  const int arow = mt * 16 + l16;                // A row for this lane
  const int bcol = nt * 16 + l16;                // B col for this lane
  const int koff = half * 2;                     // lane-half K offset

  const float* __restrict__ Arow = A + (size_t)arow * K;

  v8f acc = {};
#pragma unroll 4
  for (int k = 0; k < K; k += 4) {
    v2f a, b;
    a.x = Arow[k + koff + 0];
    a.y = Arow[k + koff + 1];
    b.x = B[(size_t)(k + koff + 0) * N + bcol];
    b.y = B[(size_t)(k + koff + 1) * N + bcol];
    acc = __builtin_amdgcn_wmma_f32_16x16x4_f32(
        /*neg_a=*/false, a, /*neg_b=*/false, b,
        /*c_mod=*/(short)0, acc, /*reuse_a=*/false, /*reuse_b=*/false);
  }

  const int crow0 = mt * 16 + half * 8;
#pragma unroll
  for (int j = 0; j < 8; ++j)
    C[(size_t)(crow0 + j) * N + bcol] = acc[j];
}

// ---------------------------------------------------------------------------
// Per-node attention scores: asrc[n,h] = sum_c xp[n,h,c]*att_src[h,c] (ditto dst)
// float4-vectorized channel loop (ch % 4 == 0).
// ---------------------------------------------------------------------------
__global__ void attn_scores(const float* __restrict__ xp,
                            const float* __restrict__ att_src,
                            const float* __restrict__ att_dst,
                            float* __restrict__ asrc,
                            float* __restrict__ adst,
                            int n_nodes, int heads, int ch) {
  int idx = blockIdx.x * blockDim.x + threadIdx.x;   // n*heads + h
  if (idx >= n_nodes * heads) return;
  int h = idx % heads;
  const float4* __restrict__ v  = (const float4*)(xp + (size_t)idx * ch);
  const float4* __restrict__ ws = (const float4*)(att_src + h * ch);
  const float4* __restrict__ wd = (const float4*)(att_dst + h * ch);
  float s = 0.f, d = 0.f;
#pragma unroll 4
  for (int c = 0; c < ch / 4; ++c) {
    float4 xv = v[c], wsv = ws[c], wdv = wd[c];
    s = fmaf(xv.x, wsv.x, s); s = fmaf(xv.y, wsv.y, s);
    s = fmaf(xv.z, wsv.z, s); s = fmaf(xv.w, wsv.w, s);
    d = fmaf(xv.x, wdv.x, d); d = fmaf(xv.y, wdv.y, d);
    d = fmaf(xv.z, wdv.z, d); d = fmaf(xv.w, wdv.w, d);
  }
  asrc[idx] = s;
  adst[idx] = d;
}

// ---------------------------------------------------------------------------
// Float atomic-max via ordered-int trick (dest initialized to -inf).
// ---------------------------------------------------------------------------
__device__ __forceinline__ void atomicMaxFloat(float* addr, float value) {
  if (value >= 0.f)
    atomicMax((int*)addr, __float_as_int(value));
  else
    atomicMin((unsigned int*)addr, __float_as_uint(value));
}

__device__ __forceinline__ void edge_src_dst(const long long* __restrict__ ei,
                                             int e, int E, int& src, int& dst) {
  if (e < E) { src = (int)ei[e]; dst = (int)ei[E + e]; }
  else       { src = dst = e - E; }                      // self loops
}

__device__ __forceinline__ int edge_dst(const long long* __restrict__ ei,
                                        int e, int E) {
  return (e < E) ? (int)ei[E + e] : (e - E);
}

// Pass 1: one thread per edge, all heads:
//   raw = leaky_relu(asrc[src]+adst[dst]); segment-max into amax[dst].
__global__ void edge_pass1(const long long* __restrict__ ei, int E, int n_nodes,
                           int heads,
                           const float* __restrict__ asrc,
                           const float* __restrict__ adst,
                           float* __restrict__ raw,
                           float* __restrict__ amax) {
  int e = blockIdx.x * blockDim.x + threadIdx.x;
  if (e >= E + n_nodes) return;
  int src, dst; edge_src_dst(ei, e, E, src, dst);
#pragma unroll 4
  for (int h = 0; h < heads; ++h) {
    float a = asrc[src * heads + h] + adst[dst * heads + h];
    a = (a >= 0.f) ? a : NEG_SLOPE * a;
    raw[e * heads + h] = a;
    atomicMaxFloat(&amax[dst * heads + h], a);
  }
}

// Pass 2: one thread per edge, all heads:
//   alpha = exp(raw - amax[dst]); segment-sum into denom[dst].
__global__ void edge_pass2(const long long* __restrict__ ei, int E, int n_nodes,
                           int heads,
                           const float* __restrict__ amax,
                           float* __restrict__ raw_alpha,
                           float* __restrict__ denom) {
  int e = blockIdx.x * blockDim.x + threadIdx.x;
  if (e >= E + n_nodes) return;
  int dst = edge_dst(ei, e, E);
#pragma unroll 4
  for (int h = 0; h < heads; ++h) {
    float al = expf(raw_alpha[e * heads + h] - amax[dst * heads + h]);
    raw_alpha[e * heads + h] = al;
    atomicAdd(&denom[dst * heads + h], al);
  }
}

// Pass 3: one thread per (edge, float4 of channels); ch % 4 == 0 so a float4
// never crosses a head boundary. out[dst,c..c+3] += xp[src,c..c+3] * coeff.
__global__ void edge_pass3(const long long* __restrict__ ei, int E, int n_nodes,
                           int heads, int ch,
                           const float* __restrict__ xp,
                           const float* __restrict__ alpha,
                           const float* __restrict__ denom,
                           float* __restrict__ out) {
  const int HC  = heads * ch;
  const int HC4 = HC >> 2;
  int idx = blockIdx.x * blockDim.x + threadIdx.x;   // e*HC4 + q
  int e = idx / HC4;
  if (e >= E + n_nodes) return;
  int q  = idx - e * HC4;
  int c0 = q << 2;
  int h  = c0 / ch;
  int src, dst; edge_src_dst(ei, e, E, src, dst);
  float coeff = alpha[e * heads + h] / (denom[dst * heads + h] + 1e-16f);
  float4 v = *(const float4*)(xp + (size_t)src * HC + c0);
  float* o = out + (size_t)dst * HC + c0;
  atomicAdd(o + 0, v.x * coeff);
  atomicAdd(o + 1, v.y * coeff);
  atomicAdd(o + 2, v.z * coeff);
  atomicAdd(o + 3, v.w * coeff);
}

// ---------------------------------------------------------------------------
// Bias (+ optional ReLU), in place, float4-vectorized (F % 4 == 0).
// ---------------------------------------------------------------------------
__global__ void bias_act(float* __restrict__ h, const float* __restrict__ bias,
                         int n, int F, int do_relu) {
  int F4 = F >> 2;
  int idx = blockIdx.x * blockDim.x + threadIdx.x;   // n*F4 groups
  if (idx >= n * F4) return;
  int c0 = (idx % F4) << 2;
  const float4 b = *(const float4*)(bias + c0);
  float4 v = ((float4*)h)[idx];
  v.x += b.x; v.y += b.y; v.z += b.z; v.w += b.w;
  if (do_relu) {
    v.x = fmaxf(v.x, 0.f); v.y = fmaxf(v.y, 0.f);
    v.z = fmaxf(v.z, 0.f); v.w = fmaxf(v.w, 0.f);
  }
  ((float4*)h)[idx] = v;
}

// ---------------------------------------------------------------------------
// Final edge classifier: out[e,:] = concat(h[src], h[dst]) @ Wc + bc
// Wc [2*HID, OUTCH] cached in LDS; one thread per edge (4 outputs each).
// ---------------------------------------------------------------------------
__global__ void edge_mlp(const long long* __restrict__ ei, int E,
                         const float* __restrict__ h,
                         const float* __restrict__ Wc,
                         const float* __restrict__ bc,
                         float* __restrict__ out) {
  __shared__ float sW[2 * HID * OUTCH];
  __shared__ float sb[OUTCH];
  for (int i = threadIdx.x; i < 2 * HID * OUTCH; i += blockDim.x) sW[i] = Wc[i];
  if (threadIdx.x < OUTCH) sb[threadIdx.x] = bc[threadIdx.x];
  __syncthreads();

  int e = blockIdx.x * blockDim.x + threadIdx.x;
  if (e >= E) return;
  int r = (int)ei[e];
  int c = (int)ei[E + e];
  const float4* __restrict__ hr = (const float4*)(h + (size_t)r * HID);
  const float4* __restrict__ hc = (const float4*)(h + (size_t)c * HID);

  float acc0 = sb[0], acc1 = sb[1], acc2 = sb[2], acc3 = sb[3];
#pragma unroll 4
  for (int k4 = 0; k4 < HID / 4; ++k4) {
    float4 v = hr[k4];
    const float* w = &sW[(k4 * 4) * OUTCH];
    acc0 = fmaf(v.x, w[0], acc0);  acc1 = fmaf(v.x, w[1], acc1);
    acc2 = fmaf(v.x, w[2], acc2);  acc3 = fmaf(v.x, w[3], acc3);
    acc0 = fmaf(v.y, w[4], acc0);  acc1 = fmaf(v.y, w[5], acc1);
    acc2 = fmaf(v.y, w[6], acc2);  acc3 = fmaf(v.y, w[7], acc3);
    acc0 = fmaf(v.z, w[8], acc0);  acc1 = fmaf(v.z, w[9], acc1);
    acc2 = fmaf(v.z, w[10], acc2); acc3 = fmaf(v.z, w[11], acc3);
    acc0 = fmaf(v.w, w[12], acc0); acc1 = fmaf(v.w, w[13], acc1);
    acc2 = fmaf(v.w, w[14], acc2); acc3 = fmaf(v.w, w[15], acc3);
  }
#pragma unroll 4
  for (int k4 = 0; k4 < HID / 4; ++k4) {
    float4 v = hc[k4];
    const float* w = &sW[(HID + k4 * 4) * OUTCH];
    acc0 = fmaf(v.x, w[0], acc0);  acc1 = fmaf(v.x, w[1], acc1);
    acc2 = fmaf(v.x, w[2], acc2);  acc3 = fmaf(v.x, w[3], acc3);
    acc0 = fmaf(v.y, w[4], acc0);  acc1 = fmaf(v.y, w[5], acc1);
    acc2 = fmaf(v.y, w[6], acc2);  acc3 = fmaf(v.y, w[7], acc3);
    acc0 = fmaf(v.z, w[8], acc0);  acc1 = fmaf(v.z, w[9], acc1);
    acc2 = fmaf(v.z, w[10], acc2); acc3 = fmaf(v.z, w[11], acc3);
    acc0 = fmaf(v.w, w[12], acc0); acc1 = fmaf(v.w, w[13], acc1);
    acc2 = fmaf(v.w, w[14], acc2); acc3 = fmaf(v.w, w[15], acc3);
  }
  float* o = out + (size_t)e * OUTCH;
  o[0] = acc0; o[1] = acc1; o[2] = acc2; o[3] = acc3;
}

// ---------------------------------------------------------------------------
// Host-side orchestration (graph-capture safe: only kernel launches on stream)
// ---------------------------------------------------------------------------
static inline int cdiv(long long a, long long b) { return (int)((a + b - 1) / b); }

extern "C" void kernel_launch(void* const* d_in, const int* in_sizes, int n_in,
                              void* d_out, int out_size, void* d_ws, size_t ws_size,
                              hipStream_t stream) {
  (void)in_sizes; (void)n_in; (void)out_size; (void)ws_size;

  const float*      x        = (const float*)d_in[0];
  const long long*  ei       = (const long long*)d_in[1];   // int64 [2, E]
  /* d_in[2] = w (edge_attr) is unused by the reference model */
  const float*      W1       = (const float*)d_in[3];
  const float*      att_src1 = (const float*)d_in[4];
  const float*      att_dst1 = (const float*)d_in[5];
  const float*      b1       = (const float*)d_in[6];
  const float*      W2       = (const float*)d_in[7];
  const float*      att_src2 = (const float*)d_in[8];
  const float*      att_dst2 = (const float*)d_in[9];
  const float*      b2       = (const float*)d_in[10];
  const float*      Wc       = (const float*)d_in[11];
  const float*      bc       = (const float*)d_in[12];
  float*            out      = (float*)d_out;

  const int Ep = NEDGES + NNODES;   // edges incl. self loops

  // ---- workspace layout ----
  size_t off = 0;
  char* base = (char*)d_ws;
  auto alloc = [&](size_t bytes) -> float* {
    float* p = (float*)(base + off);
    off += (bytes + 255) & ~(size_t)255;
    return p;
  };
  float* xp1   = alloc((size_t)NNODES * 256 * 4);  // x @ W1        [N,256]
  float* h1    = alloc((size_t)NNODES * 256 * 4);  // layer-1 out   [N,256]
  float* xp2   = alloc((size_t)NNODES * 64 * 4);   // h1 @ W2       [N,64]
  float* h2    = alloc((size_t)NNODES * 64 * 4);   // layer-2 out   [N,64]
  float* as1   = alloc((size_t)NNODES * HEADS * 4);
  float* ad1   = alloc((size_t)NNODES * HEADS * 4);
  float* mx1   = alloc((size_t)NNODES * HEADS * 4);
  float* dn1   = alloc((size_t)NNODES * HEADS * 4);
  float* as2   = alloc((size_t)NNODES * 4);
  float* ad2   = alloc((size_t)NNODES * 4);
  float* mx2   = alloc((size_t)NNODES * 4);
  float* dn2   = alloc((size_t)NNODES * 4);
  float* alpha = alloc((size_t)Ep * HEADS * 4);    // reused by layer 2 (1 head)

  const int TB = 256;
  const int FILLB = 2048;

  // ---- init accumulators ----
  fill_f32<<<FILLB, TB, 0, stream>>>(h1, 0.f, (size_t)NNODES * 256);
  fill_f32<<<FILLB, TB, 0, stream>>>(h2, 0.f, (size_t)NNODES * 64);
  fill_f32<<<FILLB, TB, 0, stream>>>(mx1, -INFINITY, (size_t)NNODES * HEADS);
  fill_f32<<<FILLB, TB, 0, stream>>>(dn1, 0.f, (size_t)NNODES * HEADS);
  fill_f32<<<FILLB, TB, 0, stream>>>(mx2, -INFINITY, (size_t)NNODES);
  fill_f32<<<FILLB, TB, 0, stream>>>(dn2, 0.f, (size_t)NNODES);

  // ======================= Layer 1: GATConv(256 -> 4x64) ====================
  {
    int waves = (NNODES / 16) * (256 / 16);                 // 50000 tiles
    wmma_gemm_f32<<<cdiv(waves, TB / 32), TB, 0, stream>>>(x, W1, xp1,
                                                           NNODES, 256, 256);
    attn_scores<<<cdiv((long long)NNODES * HEADS, TB), TB, 0, stream>>>(
        xp1, att_src1, att_dst1, as1, ad1, NNODES, HEADS, HID);
    edge_pass1<<<cdiv(Ep, TB), TB, 0, stream>>>(
        ei, NEDGES, NNODES, HEADS, as1, ad1, alpha, mx1);
    edge_pass2<<<cdiv(Ep, TB), TB, 0, stream>>>(
        ei, NEDGES, NNODES, HEADS, mx1, alpha, dn1);
    edge_pass3<<<cdiv((long long)Ep * (HEADS * HID / 4), TB), TB, 0, stream>>>(
        ei, NEDGES, NNODES, HEADS, HID, xp1, alpha, dn1, h1);
    bias_act<<<cdiv((long long)NNODES * (256 / 4), TB), TB, 0, stream>>>(
        h1, b1, NNODES, 256, /*relu=*/1);
  }

  // ======================= Layer 2: GATConv(256 -> 1x64) ====================
  {
    int waves = (NNODES / 16) * (64 / 16);                  // 12500 tiles
    wmma_gemm_f32<<<cdiv(waves, TB / 32), TB, 0, stream>>>(h1, W2, xp2,
                                                           NNODES, 64, 256);
    attn_scores<<<cdiv((long long)NNODES, TB), TB, 0, stream>>>(
        xp2, att_src2, att_dst2, as2, ad2, NNODES, 1, HID);
    edge_pass1<<<cdiv(Ep, TB), TB, 0, stream>>>(
        ei, NEDGES, NNODES, 1, as2, ad2, alpha, mx2);
    edge_pass2<<<cdiv(Ep, TB), TB, 0, stream>>>(
        ei, NEDGES, NNODES, 1, mx2, alpha, dn2);
    edge_pass3<<<cdiv((long long)Ep * (HID / 4), TB), TB, 0, stream>>>(
        ei, NEDGES, NNODES, 1, HID, xp2, alpha, dn2, h2);
    bias_act<<<cdiv((long long)NNODES * (64 / 4), TB), TB, 0, stream>>>(
        h2, b2, NNODES, 64, /*relu=*/0);
  }

  // ======================= Edge classifier head =============================
  edge_mlp<<<cdiv(NEDGES, TB), TB, 0, stream>>>(ei, NEDGES, h2, Wc, bc, out);
}